// AttentionPooling_6923487282068
// MI455X (gfx1250) — compile-verified
//
#include <hip/hip_runtime.h>

// ---------------------------------------------------------------------------
// AttentionPooling for MI455X (gfx1250, wave32).
//
// Roofline: after algebraic rewrite (precompute qk = q@Wk; pooled =
// (attn-weighted sum of h) @ Wv^T), the only large traffic is ONE streaming
// pass over gene_hiddens (819 MB -> ~35 us @ 23.3 TB/s). The streaming kernel
// is register-resident (float4/global_load_b128), uses online softmax, and is
// HBM-bound. The 640x16x640 f32 GEMM at the end uses V_WMMA_F32_16X16X4_F32.
// ---------------------------------------------------------------------------

#define HIDDEN    640
#define ATTN_DIM  128
#define BATCH     16
#define NTOK      20000
#define WGS_PER_B 40
#define TOK_PER_WG (NTOK / WGS_PER_B)     // 500
#define LN_EPS    1e-5f
#define SCORE_SCALE 0.08838834764831845f  // 1/sqrt(128)
#define J4 5                              // 640 floats / (32 lanes * 4/f4) = 5

typedef __attribute__((ext_vector_type(2))) float v2f;
typedef __attribute__((ext_vector_type(8))) float v8f;

// ---------------------------------------------------------------------------
// Kernel 1: qk[b,h] = scale * sum_a (pos[b,:]·Wq[a,:]) * Wk[a,h]   [16 x 640]
// ---------------------------------------------------------------------------
__global__ void __launch_bounds__(256)
qk_kernel(const float* __restrict__ pos, const float* __restrict__ Wq,
          const float* __restrict__ Wk, float* __restrict__ qk) {
  const int b = blockIdx.x;
  __shared__ float q_s[ATTN_DIM];
  if (threadIdx.x < ATTN_DIM) {
    float s = 0.f;
    const float* pr = pos + (size_t)b * HIDDEN;
    const float* wr = Wq + (size_t)threadIdx.x * HIDDEN;
    for (int h = 0; h < HIDDEN; ++h) s = fmaf(pr[h], wr[h], s);
    q_s[threadIdx.x] = s;
  }
  __syncthreads();
  for (int h = threadIdx.x; h < HIDDEN; h += 256) {
    float s = 0.f;
    for (int a = 0; a < ATTN_DIM; ++a) s = fmaf(q_s[a], Wk[(size_t)a * HIDDEN + h], s);
    qk[(size_t)b * HIDDEN + h] = s * SCORE_SCALE;
  }
}

// ---------------------------------------------------------------------------
// Kernel 2: fused streaming pass. Each wave keeps a token row (640 f32) in
// registers, computes score = qk·h via cross-lane reduce, and maintains an
// online-softmax accumulator acc[640] (lane-distributed, 20 VGPRs).
// Per-workgroup (m, l, acc[640]) partials are merged in LDS then written out.
// ---------------------------------------------------------------------------
__global__ void __launch_bounds__(256)
attn_stream_kernel(const float* __restrict__ gh, const float* __restrict__ qk,
                   float* __restrict__ partials) {
  const int b    = blockIdx.x / WGS_PER_B;
  const int wg   = blockIdx.x % WGS_PER_B;
  const int lane = threadIdx.x & 31;
  const int wave = threadIdx.x >> 5;

  __shared__ float sm[8];
  __shared__ float sl[8];
  __shared__ float sacc[8][HIDDEN];

  const float4* qkp = (const float4*)(qk + (size_t)b * HIDDEN);
  float4 qv[J4];
#pragma unroll
  for (int j = 0; j < J4; ++j) qv[j] = qkp[j * 32 + lane];

  float m = -3.0e38f, l = 0.f;
  float4 acc[J4];
#pragma unroll
  for (int j = 0; j < J4; ++j) acc[j] = make_float4(0.f, 0.f, 0.f, 0.f);

  const int nbase = wg * TOK_PER_WG;
  const int nend  = nbase + TOK_PER_WG;
  for (int n = nbase + wave; n < nend; n += 8) {
    const float4* hp = (const float4*)(gh + ((size_t)b * NTOK + n) * HIDDEN);
    if (n + 8 < nend)  // hint: next row for this wave (global_prefetch_b8)
      __builtin_prefetch(hp + 8 * (HIDDEN / 4) + lane, 0, 0);

    float4 hv[J4];
#pragma unroll
    for (int j = 0; j < J4; ++j) hv[j] = hp[j * 32 + lane];

    float p = 0.f;
#pragma unroll
    for (int j = 0; j < J4; ++j) {
      p = fmaf(hv[j].x, qv[j].x, p);
      p = fmaf(hv[j].y, qv[j].y, p);
      p = fmaf(hv[j].z, qv[j].z, p);
      p = fmaf(hv[j].w, qv[j].w, p);
    }
#pragma unroll
    for (int off = 16; off > 0; off >>= 1) p += __shfl_xor(p, off, 32);

    if (p > m) {  // wave-uniform branch: rescale running state
      float corr = __expf(m - p);
      m = p;
      l *= corr;
#pragma unroll
      for (int j = 0; j < J4; ++j) {
        acc[j].x *= corr; acc[j].y *= corr; acc[j].z *= corr; acc[j].w *= corr;
      }
    }
    float w = __expf(p - m);
    l += w;
#pragma unroll
    for (int j = 0; j < J4; ++j) {
      acc[j].x = fmaf(w, hv[j].x, acc[j].x);
      acc[j].y = fmaf(w, hv[j].y, acc[j].y);
      acc[j].z = fmaf(w, hv[j].z, acc[j].z);
      acc[j].w = fmaf(w, hv[j].w, acc[j].w);
    }
  }

  // -- merge 8 waves in LDS (flash-softmax combine) --
  if (lane == 0) sm[wave] = m;
  __syncthreads();
  float M = sm[0];
#pragma unroll
  for (int w2 = 1; w2 < 8; ++w2) M = fmaxf(M, sm[w2]);
  float corr = __expf(m - M);
  if (lane == 0) sl[wave] = l * corr;
  float4* sap = (float4*)&sacc[wave][0];
#pragma unroll
  for (int j = 0; j < J4; ++j) {
    float4 a = acc[j];
    a.x *= corr; a.y *= corr; a.z *= corr; a.w *= corr;
    sap[j * 32 + lane] = a;
  }
  __syncthreads();

  float* out = partials + (size_t)blockIdx.x * (HIDDEN + 2);
  if (threadIdx.x == 0) {
    float L = 0.f;
    for (int w2 = 0; w2 < 8; ++w2) L += sl[w2];
    out[0] = M;
    out[1] = L;
  }
  for (int h = threadIdx.x; h < HIDDEN; h += 256) {
    float s = 0.f;
#pragma unroll
    for (int w2 = 0; w2 < 8; ++w2) s += sacc[w2][h];
    out[2 + h] = s;
  }
}

// ---------------------------------------------------------------------------
// Kernel 3: merge WGS_PER_B partials per batch -> sD[b,h] = softmax-weighted
// mean of gene_hiddens rows (already divided by total L).
// ---------------------------------------------------------------------------
__global__ void __launch_bounds__(256)
merge_partials_kernel(const float* __restrict__ partials, float* __restrict__ sD) {
  const int b = blockIdx.x;
  const float* base = partials + (size_t)b * WGS_PER_B * (HIDDEN + 2);
  __shared__ float coef[WGS_PER_B];
  __shared__ float invLs;

  float M = -3.0e38f;
  for (int i = 0; i < WGS_PER_B; ++i) M = fmaxf(M, base[(size_t)i * (HIDDEN + 2)]);
  if (threadIdx.x < WGS_PER_B)
    coef[threadIdx.x] = __expf(base[(size_t)threadIdx.x * (HIDDEN + 2)] - M);
  __syncthreads();
  if (threadIdx.x == 0) {
    float L = 0.f;
    for (int i = 0; i < WGS_PER_B; ++i) L += coef[i] * base[(size_t)i * (HIDDEN + 2) + 1];
    invLs = 1.0f / L;
  }
  __syncthreads();
  const float invL = invLs;
  for (int h = threadIdx.x; h < HIDDEN; h += 256) {
    float s = 0.f;
    for (int i = 0; i < WGS_PER_B; ++i)
      s += coef[i] * base[(size_t)i * (HIDDEN + 2) + 2 + h];
    sD[(size_t)b * HIDDEN + h] = s * invL;
  }
}

// ---------------------------------------------------------------------------
// Kernel 4: pooled[b,v] = sum_h Wv[v,h] * sD[b,h] via V_WMMA_F32_16X16X4_F32.
// One workgroup (8 waves); wave w owns M-tiles 5w..5w+4 of the 640-row output,
// N = 16 batches, K = 640 in steps of 4.
// A-layout (16x4 f32): lane%16 = M row; VGPR0 holds K = 2*(lane>=16),
// VGPR1 holds K+1. B-layout (4x16) mirrors with N = lane%16.
// C-layout: VGPR r -> M = r + 8*(lane>=16), N = lane%16.
// ---------------------------------------------------------------------------
__global__ void __launch_bounds__(256)
pool_matmul_wmma(const float* __restrict__ Wv, const float* __restrict__ sD,
                 float* __restrict__ pooled) {
  const int lane  = threadIdx.x & 31;
  const int wave  = threadIdx.x >> 5;
  const int col   = lane & 15;    // N index = batch
  const int khalf = lane >> 4;    // 0 -> K{0,1}, 1 -> K{2,3}

  const float* brow = sD + (size_t)col * HIDDEN;
  for (int t = 0; t < 5; ++t) {
    const int vbase = (wave * 5 + t) * 16;
    const float* arow = Wv + (size_t)(vbase + col) * HIDDEN;
    v8f c = {};
    for (int k0 = 0; k0 < HIDDEN; k0 += 4) {
      v2f a, bb;
      a.x  = arow[k0 + 2 * khalf];
      a.y  = arow[k0 + 2 * khalf + 1];
      bb.x = brow[k0 + 2 * khalf];
      bb.y = brow[k0 + 2 * khalf + 1];
      c = __builtin_amdgcn_wmma_f32_16x16x4_f32(
          false, a, false, bb, (short)0, c, false, false);
    }
#pragma unroll
    for (int r = 0; r < 8; ++r) {
      const int v = vbase + r + 8 * khalf;
      pooled[(size_t)col * HIDDEN + v] = c[r];
    }
  }
}

// ---------------------------------------------------------------------------
// Kernel 5: LayerNorm over HIDDEN per batch row.
// ---------------------------------------------------------------------------
__global__ void __launch_bounds__(256)
ln_kernel(const float* __restrict__ pooled, const float* __restrict__ gamma,
          const float* __restrict__ beta, float* __restrict__ out) {
  const int b = blockIdx.x;
  const float* p = pooled + (size_t)b * HIDDEN;
  __shared__ float rs[256], rs2[256];
  float s = 0.f, s2 = 0.f;
  for (int h = threadIdx.x; h < HIDDEN; h += 256) {
    float x = p[h];
    s += x;
    s2 = fmaf(x, x, s2);
  }
  rs[threadIdx.x] = s;
  rs2[threadIdx.x] = s2;
  __syncthreads();
  for (int off = 128; off > 0; off >>= 1) {
    if (threadIdx.x < off) {
      rs[threadIdx.x]  += rs[threadIdx.x + off];
      rs2[threadIdx.x] += rs2[threadIdx.x + off];
    }
    __syncthreads();
  }
  const float mean = rs[0] * (1.0f / HIDDEN);
  const float var  = rs2[0] * (1.0f / HIDDEN) - mean * mean;
  const float rstd = rsqrtf(var + LN_EPS);
  for (int h = threadIdx.x; h < HIDDEN; h += 256) {
    out[(size_t)b * HIDDEN + h] = (p[h] - mean) * rstd * gamma[h] + beta[h];
  }
}

// ---------------------------------------------------------------------------
extern "C" void kernel_launch(void* const* d_in, const int* in_sizes, int n_in,
                              void* d_out, int out_size, void* d_ws, size_t ws_size,
                              hipStream_t stream) {
  const float* gh    = (const float*)d_in[0];  // [16,20000,640]
  const float* pos   = (const float*)d_in[1];  // [16,640]
  const float* Wq    = (const float*)d_in[2];  // [128,640]
  const float* Wk    = (const float*)d_in[3];  // [128,640]
  const float* Wv    = (const float*)d_in[4];  // [640,640]
  const float* gamma = (const float*)d_in[5];  // [640]
  const float* beta  = (const float*)d_in[6];  // [640]
  float* out = (float*)d_out;                  // [16,640]

  float* ws        = (float*)d_ws;
  float* ws_qk     = ws;                                           // 16*640
  float* ws_part   = ws_qk + (size_t)BATCH * HIDDEN;               // 640*642
  float* ws_sD     = ws_part + (size_t)BATCH * WGS_PER_B * (HIDDEN + 2);
  float* ws_pooled = ws_sD + (size_t)BATCH * HIDDEN;               // 16*640

  qk_kernel<<<BATCH, 256, 0, stream>>>(pos, Wq, Wk, ws_qk);
  attn_stream_kernel<<<BATCH * WGS_PER_B, 256, 0, stream>>>(gh, ws_qk, ws_part);
  merge_partials_kernel<<<BATCH, 256, 0, stream>>>(ws_part, ws_sD);
  pool_matmul_wmma<<<1, 256, 0, stream>>>(Wv, ws_sD, ws_pooled);
  ln_kernel<<<BATCH, 256, 0, stream>>>(ws_pooled, gamma, beta, out);
}